// GSTE_88622355186258
// MI455X (gfx1250) — compile-verified
//
#include <hip/hip_runtime.h>

// GSTE quantizer, elementwise over 4096x8192 fp32.
// Roofline: 403 MB streamed @ 23.3 TB/s ~= 17.3 us floor (HBM-bound).
// Round-1 version was VALU-bound (~80 valu/elem from 2x16 select chains).
// This version uses closed-form bucketization (~21 valu/elem) when a[] is
// uniformly spaced (true for the given inputs; verified per-wave via scalar
// loads -> s_cbranch), with the general select-chain path as fallback.

typedef __attribute__((ext_vector_type(4))) float v4f;

#define NV        16        // number of quant levels (2^4)
#define INTERVAL  0.125f    // 2.0 / 16, Python-float constant in the reference
#define TPB       256       // 8 waves per block (wave32)
#define TILE      8         // float4s per thread -> 32 elements per thread

__global__ __launch_bounds__(TPB) void gste_kernel(
    const v4f* __restrict__ xv,
    const v4f* __restrict__ mv,
    const float* __restrict__ start_p,
    const float* __restrict__ a_p,
    const float* __restrict__ scale1_p,
    v4f* __restrict__ ov,
    int n4)
{
    // ---- Wave-uniform setup (scalar loads -> SGPRs) ------------------------
    const float s0 = start_p[0];
    const float s1 = scale1_p[0];
    const float a0 = a_p[0];

    bool uniform = true;
#pragma unroll
    for (int k = 1; k < NV; ++k) uniform = uniform && (a_p[k] == a0);

    const int base = blockIdx.x * (TPB * TILE) + threadIdx.x;

    if (uniform) {
        // ==== Fast path: thresholds affine in k =============================
        // tf[k] = tf0 + A*k, tb[k] = tb0 + A*k
        // idx  = #{thre < x} = clamp(ceil((x - t0)/A), 0, 16); idx>0 <=> x>t0
        const float A    = a0;
        const float invA = 1.0f / A;            // exact for A = 0.125
        const float tf0  = s0 + 0.5f * A;
        const float tb0  = s0;
        const float bf   = -tf0 * invA;         // uf = fma(x, invA, bf)
        const float bb   = -tb0 * invA;
        const float c0   = INTERVAL * invA;     // INTERVAL/a[j], == 1 at init

#pragma unroll
        for (int t = 0; t < TILE; ++t) {
            int i = base + t * TPB;
            if (i < n4) {
                v4f x4 = __builtin_nontemporal_load(xv + i);
                v4f m4 = __builtin_nontemporal_load(mv + i);
                v4f o4;
#pragma unroll
                for (int c = 0; c < 4; ++c) {
                    float x = x4[c];

                    // forward: xf = idx_f*INTERVAL if idx_f>0 else x*scale1
                    float uf = __builtin_fmaf(x, invA, bf);
                    float cf = fminf(__builtin_ceilf(uf), 16.0f);
                    float xf = (x > tf0) ? cf * INTERVAL : x * s1;

                    // backward: j = clip(idx_b-1, 0, 15);
                    // xb = c0*(x - tb[j]) + j*INTERVAL if idx_b>0 else x
                    float ub  = __builtin_fmaf(x, invA, bb);
                    float jf  = fminf(fmaxf(__builtin_ceilf(ub) - 1.0f, 0.0f), 15.0f);
                    float tbj = __builtin_fmaf(jf, A, tb0);
                    float xb  = (x > tb0)
                                  ? __builtin_fmaf(c0, x - tbj, jf * INTERVAL)
                                  : x;

                    // reference value rounding: (stop_grad(xf)+xb-stop_grad(xb))*mask
                    float o = (xf + xb) - xb;
                    o4[c] = o * m4[c];
                }
                __builtin_nontemporal_store(o4, ov + i);
            }
        }
    } else {
        // ==== General path: arbitrary a[] -> 16-way select chains ===========
        float tf[NV], tb[NV], c0[NV], c1[NV];
        {
            float h_prev = a0 * 0.5f;
            tf[0] = s0 + h_prev;
            tb[0] = s0;
            c0[0] = INTERVAL / a0;
            c1[0] = 0.0f - c0[0] * tb[0];
#pragma unroll
            for (int k = 1; k < NV; ++k) {
                float ak = a_p[k];
                float h  = ak * 0.5f;
                tf[k] = tf[k - 1] + h + h_prev;
                h_prev = h;
                tb[k] = tb[k - 1] + a_p[k - 1];
                c0[k] = INTERVAL / ak;
                c1[k] = (float)k * INTERVAL - c0[k] * tb[k];
            }
        }

#pragma unroll
        for (int t = 0; t < TILE; ++t) {
            int i = base + t * TPB;
            if (i < n4) {
                v4f x4 = __builtin_nontemporal_load(xv + i);
                v4f m4 = __builtin_nontemporal_load(mv + i);
                v4f o4;
#pragma unroll
                for (int c = 0; c < 4; ++c) {
                    float x  = x4[c];
                    float xf = x * s1;
                    float xb = x;
#pragma unroll
                    for (int k = 0; k < NV; ++k) {
                        if (x > tf[k]) xf = (float)(k + 1) * INTERVAL;
                        if (x > tb[k]) xb = __builtin_fmaf(c0[k], x, c1[k]);
                    }
                    float o = (xf + xb) - xb;
                    o4[c] = o * m4[c];
                }
                __builtin_nontemporal_store(o4, ov + i);
            }
        }
    }
}

extern "C" void kernel_launch(void* const* d_in, const int* in_sizes, int n_in,
                              void* d_out, int out_size, void* d_ws, size_t ws_size,
                              hipStream_t stream) {
    // setup_inputs() order: x, mask, start, a, scale1
    const v4f*   x      = (const v4f*)d_in[0];
    const v4f*   mask   = (const v4f*)d_in[1];
    const float* start  = (const float*)d_in[2];
    const float* a      = (const float*)d_in[3];
    const float* scale1 = (const float*)d_in[4];
    v4f*         out    = (v4f*)d_out;

    int n  = in_sizes[0];          // 4096*8192, divisible by 4
    int n4 = n >> 2;               // 8,388,608 float4s
    int per_block = TPB * TILE;    // 2048 float4s per block
    int blocks = (n4 + per_block - 1) / per_block;  // 4096 blocks

    gste_kernel<<<blocks, TPB, 0, stream>>>(x, mask, start, a, scale1, out, n4);
}